// GCN_20761871909282
// MI455X (gfx1250) — compile-verified
//
#include <hip/hip_runtime.h>

typedef __attribute__((ext_vector_type(2))) float v2f;
typedef __attribute__((ext_vector_type(4))) float v4f;
typedef __attribute__((ext_vector_type(8))) float v8f;

#define NDIM   1024   // adj is (b, NDIM, NDIM)
#define FDIM   128    // feature dim
#define TILE_M 64
#define TILE_K 32
#define APAD   36     // A LDS row stride (dwords): 144B rows -> 16B-aligned async chunks, conflict-free frag reads
#define BPAD   34     // B^T LDS row stride (dwords): even (b64 align), conflict-free frag reads, 4-way store conflicts only

// Async global->LDS copy path (gfx1250): bypasses VGPRs, tracked by ASYNCcnt.
#if defined(__HIP_DEVICE_COMPILE__) && defined(__gfx1250__) && \
    __has_builtin(__builtin_amdgcn_global_load_async_to_lds_b128)
  #define USE_ASYNC 1
#else
  #define USE_ASYNC 0
#endif

#if !USE_ASYNC
  #define WAIT_ASYNCCNT(n)
#elif __has_builtin(__builtin_amdgcn_s_wait_asynccnt)
  #define WAIT_ASYNCCNT(n) __builtin_amdgcn_s_wait_asynccnt(n)
#else
  #define WAIT_ASYNCCNT(n) asm volatile("s_wait_asynccnt %0" :: "i"(n) : "memory")
#endif

// global_load_async_to_lds_b128 signature: (AS1 v4i*, AS3 v4i*, imm offset, imm cpol)
typedef int v4i_ __attribute__((vector_size(16)));
typedef __attribute__((address_space(1))) v4i_ gv4i;
typedef __attribute__((address_space(3))) v4i_ lv4i;

// ---- A tile (adj, the 128MB HBM-dominant stream): async copy, 2 b128 per wave.
__device__ __forceinline__ void issue_A_async(
    const float* __restrict__ adjB, int m0, int kc, float* Ab, int tid)
{
#if USE_ASYNC
#pragma unroll
    for (int i = 0; i < 2; ++i) {
        const int c = tid + i * 256;
        const int row = c >> 3, cq = c & 7;
        const float* gp = adjB + (size_t)(m0 + row) * NDIM + kc + cq * 4;
        __builtin_amdgcn_global_load_async_to_lds_b128(
            (gv4i*)const_cast<float*>(gp),
            (lv4i*)(Ab + row * APAD + cq * 4), 0, 0);
    }
#else
#pragma unroll
    for (int i = 0; i < 2; ++i) {
        const int c = tid + i * 256;
        const int row = c >> 3, cq = c & 7;
        v4f v = *(const v4f*)(adjB + (size_t)(m0 + row) * NDIM + kc + cq * 4);
        *(v4f*)(Ab + row * APAD + cq * 4) = v;
    }
#endif
}

// ---- B tile (rm, L2-resident): software-pipelined through VGPRs with transpose,
//      so fragments become a single aligned ds_load_b64.
__device__ __forceinline__ void load_B_regs(
    const float* __restrict__ rmB, int kc, int tid, v4f breg[4])
{
#pragma unroll
    for (int i = 0; i < 4; ++i) {
        const int c = tid + i * 256;
        const int k = c >> 5, nq = c & 31;
        breg[i] = *(const v4f*)(rmB + (size_t)(kc + k) * FDIM + nq * 4);
    }
}

__device__ __forceinline__ void store_B_lds(float* Bt, int tid, const v4f breg[4])
{
#pragma unroll
    for (int i = 0; i < 4; ++i) {
        const int c = tid + i * 256;
        const int k = c >> 5, n = (c & 31) * 4;
#pragma unroll
        for (int j = 0; j < 4; ++j)
            Bt[(n + j) * BPAD + k] = breg[i][j];   // pairs into ds_store_2addr_b32
    }
}

// C[b,i,f] = sum_j adj[b,i,j]*rm[b,j,f]  -  adj[b,i,i]*rm[b,i,f]
// Block: 256 threads = 8 waves, 64(M) x 128(N) tile, K double-buffered in LDS.
__global__ __launch_bounds__(256) void gcn_spmm_wmma_kernel(
    const float* __restrict__ rm,    // (32, 1024, 128)
    const float* __restrict__ adj,   // (32, 1024, 1024)
    float* __restrict__ out)         // (32, 1024, 128)
{
    __shared__ float Asm[2][TILE_M * APAD];   // A: [m][k]
    __shared__ float Bsm[2][FDIM * BPAD];     // B^T: [n][k]

    const int b    = blockIdx.y;
    const int m0   = blockIdx.x * TILE_M;
    const int tid  = threadIdx.x;
    const int wave = tid >> 5;               // wave32
    const int lane = tid & 31;
    const int ln15 = lane & 15;

    const float* __restrict__ adjB = adj + (size_t)b * NDIM * NDIM;
    const float* __restrict__ rmB  = rm  + (size_t)b * NDIM * FDIM;
    float* __restrict__ outB       = out + (size_t)b * NDIM * FDIM;

    const int wm16 = wave >> 1;              // 0..3  : 16-row strip
    const int wn0  = (wave & 1) * 64;        // 0/64  : 64-col half

    v8f acc[4];
#pragma unroll
    for (int t = 0; t < 4; ++t)
#pragma unroll
        for (int i = 0; i < 8; ++i) acc[t][i] = 0.0f;

    v4f breg[4];

    // Prologue: chunk 0 -> buffer 0
    issue_A_async(adjB, m0, 0, Asm[0], tid);
    load_B_regs(rmB, 0, tid, breg);
    store_B_lds(Bsm[0], tid, breg);

    for (int kc = 0; kc < NDIM; kc += TILE_K) {
        const int cur = (kc >> 5) & 1;

        if (kc + TILE_K < NDIM) {
            // Next chunk: A straight to LDS (async), B into registers (stored after compute).
            issue_A_async(adjB, m0, kc + TILE_K, Asm[cur ^ 1], tid);
            load_B_regs(rmB, kc + TILE_K, tid, breg);
            WAIT_ASYNCCNT(2);    // this chunk's 2 A-asyncs retired
        } else {
            WAIT_ASYNCCNT(0);
        }
        __syncthreads();         // all waves' LDS writes for buf[cur] visible

        const float* Ab = Asm[cur];
        const float* Bt = Bsm[cur];

        // 8 steps of K=4. A 16x4 f32 layout: lanes 0-15 -> K=kk,kk+1;
        // lanes 16-31 -> K=kk+2,kk+3; B mirrors with N = lane&15.
#pragma unroll
        for (int ks = 0; ks < 8; ++ks) {
            const int kk = ks * 4 + ((lane >> 4) << 1);
            const int am = (wm16 << 4) + ln15;
            v2f afrag = *(const v2f*)&Ab[am * APAD + kk];
#pragma unroll
            for (int nt = 0; nt < 4; ++nt) {
                const int n = wn0 + (nt << 4) + ln15;
                v2f bfrag = *(const v2f*)&Bt[n * BPAD + kk];   // single ds_load_b64
                acc[nt] = __builtin_amdgcn_wmma_f32_16x16x4_f32(
                    false, afrag, false, bfrag,
                    (short)0, acc[nt], false, false);
            }
        }

        if (kc + TILE_K < NDIM)
            store_B_lds(Bsm[cur ^ 1], tid, breg);   // B load latency hidden by compute

        __syncthreads();         // buf[cur] free; next-chunk B stores visible
    }

    // Epilogue: subtract diagonal term adj[i,i]*rm[i,n] (L2 hits), then store.
    // C/D layout: VGPR r -> M=r (lanes 0-15) / r+8 (lanes 16-31), N = lane&15.
    const int mbase = m0 + (wm16 << 4) + ((lane >> 4) << 3);
#pragma unroll
    for (int r = 0; r < 8; ++r) {
        const int mrow = mbase + r;
        const float d = adjB[(size_t)mrow * NDIM + mrow];
#pragma unroll
        for (int nt = 0; nt < 4; ++nt) {
            const int ncol = wn0 + (nt << 4) + ln15;
            acc[nt][r] -= d * rmB[(size_t)mrow * FDIM + ncol];
        }
    }
#pragma unroll
    for (int nt = 0; nt < 4; ++nt) {
        const int ncol = wn0 + (nt << 4) + ln15;
#pragma unroll
        for (int r = 0; r < 8; ++r)
            outB[(size_t)(mbase + r) * FDIM + ncol] = acc[nt][r];
    }
}

extern "C" void kernel_launch(void* const* d_in, const int* in_sizes, int n_in,
                              void* d_out, int out_size, void* d_ws, size_t ws_size,
                              hipStream_t stream) {
    (void)in_sizes; (void)n_in; (void)out_size; (void)d_ws; (void)ws_size;
    const float* rm  = (const float*)d_in[0];   // regional_means (32,1024,128)
    const float* adj = (const float*)d_in[1];   // adj            (32,1024,1024)
    float* out = (float*)d_out;                 // (32,1024,128)

    dim3 grid(NDIM / TILE_M, 32);   // 16 M-tiles x 32 batches = 512 blocks
    dim3 block(256);                // 8 waves (wave32)
    gcn_spmm_wmma_kernel<<<grid, block, 0, stream>>>(rm, adj, out);
}